// VRTRPostProcess_62654982914945
// MI455X (gfx1250) — compile-verified
//
#include <hip/hip_runtime.h>
#include <hip/hip_bf16.h>
#include <math.h>

// ---------------- problem constants ----------------
#define B_    8
#define Q_    192
#define P_    128
#define NACT  117
#define NCLS  92
#define QP1   193           // Q+1
#define AST   132           // padded LDS stride (breaks 16-way bank conflicts)

typedef float v2f __attribute__((ext_vector_type(2)));
typedef float v8f __attribute__((ext_vector_type(8)));

// =====================================================================
// Kernel 1: zero-fill the pair_score region (139 MB) using CDNA5 async
// LDS->global stores (GLOBAL_STORE_ASYNC_FROM_LDS_B128, ASYNCcnt path).
// Zero 4KB of LDS once, then stream 16B/lane stores from it.
// =====================================================================
__global__ __launch_bounds__(256) void zero_async_kernel(float* __restrict__ out, long n4)
{
    __shared__ float zbuf[256 * 4];     // 4 KB of zeros
    const int tid = threadIdx.x;
    reinterpret_cast<float4*>(zbuf)[tid] = make_float4(0.f, 0.f, 0.f, 0.f);
    __syncthreads();

    // LDS byte address of this lane's 16B zero chunk (low 32 bits of the
    // generic pointer are the LDS offset on gfx1250).
    unsigned lds_addr = (unsigned)(size_t)(&zbuf[tid * 4]);
    unsigned long long base = (unsigned long long)out;

    long i      = (long)blockIdx.x * blockDim.x + tid;
    long stride = (long)gridDim.x * blockDim.x;
    for (; i < n4; i += stride) {
        unsigned voff = (unsigned)(i * 16);   // byte offset, fits in 31 bits
        asm volatile(
            "global_store_async_from_lds_b128 %0, %1, %2"
            :
            : "v"(voff), "v"(lds_addr), "s"(base)
            : "memory");
    }
    asm volatile("s_wait_asynccnt 0x0" ::: "memory");
}

// =====================================================================
// Kernel 2: one workgroup (256 threads = 8 wave32) per image.
//   Phase 0: sigmoids -> acts/ms/keys (threads 0..127)
//            softmax scores/labels + boxes + masks (threads 128..255)
//   Phase 1: per-key winner detection, build A matrix in LDS
//   Phase 2: D = A x Acts via V_WMMA_F32_16X16X4_F32 (f32-exact),
//            predicated scatter of winner rows with masks folded in.
// =====================================================================
__global__ __launch_bounds__(256) void vrtr_main_kernel(
    const float* __restrict__ logits,      // (B,Q,NCLS)
    const float* __restrict__ pboxes,      // (B,Q,4)
    const float* __restrict__ pacts,       // (B,P,NACT+1)
    const float* __restrict__ pexists,     // (B,P)
    const int*   __restrict__ pairs,       // (B,P,2)
    const float* __restrict__ tsizes,      // (B,2)  [h,w]
    float* __restrict__ out)
{
    const int b   = blockIdx.x;
    const int tid = threadIdx.x;

    extern __shared__ float sm[];
    float* sActs = sm;                        // P_ * AST   (acts, padded cols zeroed)
    float* sA    = sActs + P_ * AST;          // P_ * AST   (aggregation matrix)
    float* sMs   = sA   + P_ * AST;           // P_
    float* sHm   = sMs  + P_;                 // Q_
    float* sOm   = sHm  + Q_;                 // QP1
    int*   sKey  = (int*)(sOm + QP1);         // P_
    int*   sWin  = sKey + P_;                 // P_
    int*   sH    = sWin + P_;                 // P_
    int*   sO    = sH   + P_;                 // P_

    float* out_ps     = out;
    float* out_scores = out + (size_t)B_ * NACT * Q_ * QP1;
    float* out_labels = out_scores + B_ * Q_;
    float* out_boxes  = out_labels + B_ * Q_;

    // ---------------- Phase 0 ----------------
    if (tid < P_) {
        const int p = tid;
        const float* ap = pacts + ((size_t)b * P_ + p) * (NACT + 1);
        __builtin_prefetch(ap, 0, 0);                  // global_prefetch_b8
        float e = 1.f / (1.f + expf(-pexists[b * P_ + p]));
        for (int a = 0; a < NACT; ++a)
            sActs[p * AST + a] = e / (1.f + expf(-ap[a]));
        for (int a = NACT; a < AST; ++a)
            sActs[p * AST + a] = 0.f;                  // zero pad cols
        sMs[p] = 1.f - e / (1.f + expf(-ap[NACT]));

        int h  = pairs[((size_t)b * P_ + p) * 2 + 0];
        int o  = pairs[((size_t)b * P_ + p) * 2 + 1];
        int oe = (h == o) ? Q_ : o;
        sH[p] = h; sO[p] = oe; sKey[p] = h * QP1 + oe;
    } else {
        const int t = tid - 128;
        if (t == 0) sOm[Q_] = 1.f;                     // o_ext extra column
        const float H = tsizes[b * 2 + 0];
        const float W = tsizes[b * 2 + 1];
        // 192 queries over 128 threads: q = t and (t<64) q = t+128
        for (int rep = 0; rep < 2; ++rep) {
            int q = t + rep * 128;
            if (q >= Q_) break;
            const float* lp = logits + ((size_t)b * Q_ + q) * NCLS;
            float m = -INFINITY;
            for (int c = 0; c < NCLS; ++c) m = fmaxf(m, lp[c]);
            float s = 0.f, bv = -INFINITY; int bl = 0;
            for (int c = 0; c < NCLS; ++c) {
                float v = lp[c];
                s += expf(v - m);
                if (c < NCLS - 1 && v > bv) { bv = v; bl = c; }
            }
            float score = expf(bv - m) / s;
            out_scores[b * Q_ + q] = score;
            out_labels[b * Q_ + q] = (float)bl;
            sHm[q] = (bl == 1 && score > 0.f) ? 1.f : 0.f;
            sOm[q] = (score > 0.f) ? 1.f : 0.f;

            const float* bx = pboxes + ((size_t)b * Q_ + q) * 4;
            float cx = bx[0], cy = bx[1], w = bx[2], h2 = bx[3];
            float* ob = out_boxes + ((size_t)b * Q_ + q) * 4;
            ob[0] = (cx - 0.5f * w)  * W;
            ob[1] = (cy - 0.5f * h2) * H;
            ob[2] = (cx + 0.5f * w)  * W;
            ob[3] = (cy + 0.5f * h2) * H;
        }
    }
    __syncthreads();

    // ---------------- Phase 1: winners + A matrix ----------------
    if (tid < P_) {
        const int p = tid;
        const float mp = sMs[p];
        const int   kp = sKey[p];
        int win = 1;
        for (int p2 = 0; p2 < P_; ++p2) {
            if (p2 != p && sKey[p2] == kp) {
                float m2 = sMs[p2];
                if (m2 > mp || (m2 == mp && p2 < p)) { win = 0; break; }
            }
        }
        sWin[p] = win;
    }
    __syncthreads();

    for (int idx = tid; idx < P_ * P_; idx += 256) {
        int w  = idx >> 7;
        int p2 = idx & (P_ - 1);
        float v = 0.f;
        if (sWin[w] && sKey[p2] == sKey[w])
            v = sMs[p2] * ((p2 == w) ? 2.f : 1.f);   // ssum + argmax-doubling
        sA[w * AST + p2] = v;
    }
    __syncthreads();

    // ---------------- Phase 2: WMMA GEMM + masked scatter ----------------
    const int wave = tid >> 5;           // 0..7  -> M tile
    const int lane = tid & 31;
    const int mt   = wave;
    const int rowA = mt * 16 + (lane & 15);
    const int koff = 2 * (lane >> 4);    // A/B K-split across half-waves
    const int ncol = lane & 15;

    for (int nt = 0; nt < 8; ++nt) {
        v8f c = {};
        const int col = nt * 16 + ncol;
        for (int k0 = 0; k0 < P_; k0 += 4) {
            const int kk = k0 + koff;
            v2f a, bv;
            a.x  = sA[rowA * AST + kk];
            a.y  = sA[rowA * AST + kk + 1];
            bv.x = sActs[kk * AST + col];
            bv.y = sActs[(kk + 1) * AST + col];
            c = __builtin_amdgcn_wmma_f32_16x16x4_f32(
                    false, a, false, bv, (short)0, c, false, false);
        }
        // store only winner rows (one winner per key -> race-free)
        if (col < NACT) {
            const int rbase = mt * 16 + 8 * (lane >> 4);
            for (int v = 0; v < 8; ++v) {
                const int r = rbase + v;
                if (sWin[r]) {
                    const int h = sH[r];
                    const int o = sO[r];
                    const float val = c[v] * sHm[h] * sOm[o];
                    out_ps[(((size_t)b * NACT + col) * Q_ + h) * QP1 + o] = val;
                }
            }
        }
    }
}

// =====================================================================
extern "C" void kernel_launch(void* const* d_in, const int* in_sizes, int n_in,
                              void* d_out, int out_size, void* d_ws, size_t ws_size,
                              hipStream_t stream) {
    (void)in_sizes; (void)n_in; (void)out_size; (void)d_ws; (void)ws_size;
    const float* logits = (const float*)d_in[0];
    const float* boxes  = (const float*)d_in[1];
    const float* acts   = (const float*)d_in[2];
    const float* exists = (const float*)d_in[3];
    const int*   pairs  = (const int*)d_in[4];
    const float* tsz    = (const float*)d_in[5];
    float* out = (float*)d_out;

    const long n4 = (long)B_ * NACT * Q_ * QP1 / 4;   // float4 count (exact)
    hipLaunchKernelGGL(zero_async_kernel, dim3(2048), dim3(256), 0, stream, out, n4);

    const size_t ldsBytes =
        (size_t)(2 * P_ * AST + P_ + Q_ + QP1) * sizeof(float) +
        (size_t)(4 * P_) * sizeof(int);
    hipLaunchKernelGGL(vrtr_main_kernel, dim3(B_), dim3(256), ldsBytes, stream,
                       logits, boxes, acts, exists, pairs, tsz, out);
}